// CustomMLPLayer_89378269430144
// MI455X (gfx1250) — compile-verified
//
#include <hip/hip_runtime.h>

// gathered GEMM: out[b, r] = sum_k x[b,k] * weight[indices[r], k]
// M=32 (batch), N=4403 (gathered rows), K=4096, fp32 via V_WMMA_F32_16X16X4_F32.
// HBM-bound: 72 MB gathered weight stream @ 23.3 TB/s ~= 3.1 us floor.

typedef __attribute__((ext_vector_type(2))) float v2f;
typedef __attribute__((ext_vector_type(8))) float v8f;

#define D_MODEL  4096
#define D_FF     11008
#define REMAINED 4403
#define BATCH    32

#define WAVES_PER_BLOCK 8
#define K_CHUNK (D_MODEL / WAVES_PER_BLOCK) // 512

__global__ __launch_bounds__(32 * WAVES_PER_BLOCK)
void gathered_gemm_wmma_f32(const float* __restrict__ x,       // [32][4096]
                            const float* __restrict__ weight,  // [11008][4096]
                            const int*   __restrict__ indices, // [4403]
                            float*       __restrict__ out)     // [32][4403]
{
    __shared__ float red[WAVES_PER_BLOCK][2][8][32]; // 16 KB

    const int tile = blockIdx.x;            // N columns tile*16 .. tile*16+15
    const int lane = threadIdx.x & 31;
    const int wave = threadIdx.x >> 5;

    // f32 WMMA fragment geometry (ISA 7.12.2, 32-bit A 16x4 / B 4x16):
    // lane L = (n, khalf): holds K = kb + 2*khalf + {0,1} for row/col n.
    const int n     = lane & 15;
    const int khalf = lane >> 4;

    // Gathered weight row for this lane's N column (clamped for partial tile;
    // branchless -> EXEC stays all-1s for WMMA).
    int r = tile * 16 + n;
    int rc = r < REMAINED ? r : (REMAINED - 1);
    const float* wrow  = weight + (size_t)indices[rc] * D_MODEL;
    const float* xrow0 = x + (size_t)n        * D_MODEL;  // batches 0..15
    const float* xrow1 = x + (size_t)(n + 16) * D_MODEL;  // batches 16..31

    v8f c0 = {}; // M = 0..15  x N tile
    v8f c1 = {}; // M = 16..31 x N tile

    const int kbase = wave * K_CHUNK;
    const int kend  = kbase + K_CHUNK;

    // Steady state per 16 K: 1 prefetch, 12 b64 loads, 8 WMMAs, no branches
    // besides the backedge. Unconditional speculative prefetch: an address past
    // the last mapped page is silently dropped (ISA 10.5), so no guard needed.
    for (int kb = kbase; kb < kend; kb += 16) {
        const int ko = kb + 2 * khalf;
        __builtin_prefetch(wrow + ko + 80, 0, 0); // weight stream ~320B ahead
#pragma unroll
        for (int u = 0; u < 4; ++u) {
            const int kk = ko + 4 * u;
            v2f a0 = *(const v2f*)(xrow0 + kk);
            v2f a1 = *(const v2f*)(xrow1 + kk);
            v2f b  = *(const v2f*)(wrow  + kk);
            c0 = __builtin_amdgcn_wmma_f32_16x16x4_f32(false, a0, false, b,
                                                       (short)0, c0, false, false);
            c1 = __builtin_amdgcn_wmma_f32_16x16x4_f32(false, a1, false, b,
                                                       (short)0, c1, false, false);
        }
    }

    // Spill per-wave partial accumulators to LDS for the cross-wave K reduce.
#pragma unroll
    for (int j = 0; j < 8; ++j) {
        red[wave][0][j][lane] = c0[j];
        red[wave][1][j][lane] = c1[j];
    }
    __syncthreads();

    // 2 tiles * 8 vgprs * 32 lanes = 512 outputs; 256 threads -> 2 each.
    // C/D layout: vgpr j, lane L -> M = j + 8*(L>>4) (+16 for tile 1), N = L&15.
    for (int e = threadIdx.x; e < 512; e += blockDim.x) {
        const int l = e & 31;
        const int j = (e >> 5) & 7;
        const int t = e >> 8;
        float s = 0.f;
#pragma unroll
        for (int w = 0; w < WAVES_PER_BLOCK; ++w) s += red[w][t][j][l];
        const int m  = j + 8 * (l >> 4) + 16 * t;   // batch
        const int rr = tile * 16 + (l & 15);        // gathered-row column
        if (rr < REMAINED) out[(size_t)m * REMAINED + rr] = s;
    }
}

extern "C" void kernel_launch(void* const* d_in, const int* in_sizes, int n_in,
                              void* d_out, int out_size, void* d_ws, size_t ws_size,
                              hipStream_t stream) {
    const float* x       = (const float*)d_in[0];
    const float* weight  = (const float*)d_in[1];
    const int*   indices = (const int*)d_in[2];
    float*       out     = (float*)d_out;

    const int n_tiles = (REMAINED + 15) / 16; // 276
    gathered_gemm_wmma_f32<<<n_tiles, 32 * WAVES_PER_BLOCK, 0, stream>>>(
        x, weight, indices, out);
}